// Encoder_17231408791765
// MI455X (gfx1250) — compile-verified
//
#include <hip/hip_runtime.h>
#include <hip/hip_bf16.h>

// ---------------------------------------------------------------------------
// 3-layer GAT (GATConv w/ edge features) for MI455X (gfx1250, wave32).
// Dense projections: V_WMMA_F32_16X16X4_F32, one wave owns a 16x64 strip
// (4 C tiles) so the A fragment is loaded once per K-step and reused by
// 4 consecutive WMMAs.
// Edge softmax/aggregation: 3 streaming passes with atomics.
// ---------------------------------------------------------------------------

#define Nn_NODES 50000
#define F_IN     128
#define Hh       3
#define Cc       64
#define Ee       800000
#define OUTF     64
#define FHID     (Hh * Cc)   // 192

typedef float v2f __attribute__((ext_vector_type(2)));
typedef float v8f __attribute__((ext_vector_type(8)));

// ---------------- ordered-float <-> uint monotone mapping (for atomic max) --
__device__ __forceinline__ unsigned fp_order(float f) {
    unsigned u = __float_as_uint(f);
    return (u & 0x80000000u) ? ~u : (u | 0x80000000u);
}
__device__ __forceinline__ float fp_unorder(unsigned u) {
    return (u & 0x80000000u) ? __uint_as_float(u & 0x7FFFFFFFu)
                             : __uint_as_float(~u);
}

// ---------------- fp32 GEMM via WMMA 16x16x4, 16x64 strip per wave ----------
// D[M,Nout] = A[M,K] @ B[K,Nout], all row-major fp32.
// Requires: M%16==0, K%4==0, Nout%64==0.
__global__ __launch_bounds__(32) void gemm_f32_wmma(
    const float* __restrict__ A, const float* __restrict__ B,
    float* __restrict__ D, int M, int K, int Nout)
{
    const int lane = threadIdx.x & 31;
    const int lm   = lane & 15;
    const int hi   = lane >> 4;

    const int mt = blockIdx.x;               // 16-row tile index
    const int n0 = blockIdx.y * 64;          // 64-col strip base

    const float* Arow = A + (size_t)(mt * 16 + lm) * K;  // this lane's A row
    const float* Bb   = B + n0 + lm;                     // this lane's B col base

    v8f acc0 = {}, acc1 = {}, acc2 = {}, acc3 = {};
    for (int k0 = 0; k0 < K; k0 += 4) {
        const int ka = k0 + 2 * hi;
        v2f a;
        a.x = Arow[ka];
        a.y = Arow[ka + 1];
        const float* Br0 = Bb + (size_t)ka * Nout;
        const float* Br1 = Bb + (size_t)(ka + 1) * Nout;
        v2f b0, b1, b2, b3;
        b0.x = Br0[0];  b0.y = Br1[0];
        b1.x = Br0[16]; b1.y = Br1[16];
        b2.x = Br0[32]; b2.y = Br1[32];
        b3.x = Br0[48]; b3.y = Br1[48];
        acc0 = __builtin_amdgcn_wmma_f32_16x16x4_f32(false, a, false, b0, (short)0, acc0, false, false);
        acc1 = __builtin_amdgcn_wmma_f32_16x16x4_f32(false, a, false, b1, (short)0, acc1, false, false);
        acc2 = __builtin_amdgcn_wmma_f32_16x16x4_f32(false, a, false, b2, (short)0, acc2, false, false);
        acc3 = __builtin_amdgcn_wmma_f32_16x16x4_f32(false, a, false, b3, (short)0, acc3, false, false);
    }

    float* Dcol = D + n0 + lm;
#pragma unroll
    for (int r = 0; r < 8; ++r) {
        const size_t row = (size_t)(mt * 16 + r + 8 * hi) * Nout;
        Dcol[row +  0] = acc0[r];
        Dcol[row + 16] = acc1[r];
        Dcol[row + 32] = acc2[r];
        Dcol[row + 48] = acc3[r];
    }
}

// ---------------- per-layer edge scalar: ce[h] = dot(We[h], Ae[h]) ----------
__global__ void ce_kernel(const float* __restrict__ w1e, const float* __restrict__ a1e,
                          const float* __restrict__ w2e, const float* __restrict__ a2e,
                          const float* __restrict__ w3e, const float* __restrict__ a3e,
                          float* __restrict__ ce)
{
    int t = threadIdx.x;            // 0..6
    if (t >= 7) return;
    const float *w, *a;
    if (t < 3)      { w = w1e + t * Cc;       a = a1e + t * Cc; }
    else if (t < 6) { w = w2e + (t - 3) * Cc; a = a2e + (t - 3) * Cc; }
    else            { w = w3e;                a = a3e; }
    float s = 0.f;
    for (int c = 0; c < Cc; ++c) s += w[c] * a[c];
    ce[t] = s;
}

// ---------------- per-node attention dots -----------------------------------
// asrc[n,h] = dot(hs[n, h*C:...], As[h]);  adst[n,h] = dot(hd[...], Ad[h])
__global__ void node_dots(const float* __restrict__ hs, const float* __restrict__ hd,
                          const float* __restrict__ As, const float* __restrict__ Ad,
                          float* __restrict__ asrc, float* __restrict__ adst,
                          int n_nodes, int heads)
{
    int t = blockIdx.x * blockDim.x + threadIdx.x;
    if (t >= n_nodes * heads) return;
    int n = t / heads, h = t % heads;
    const float* hr = hs + (size_t)n * heads * Cc + h * Cc;
    const float* dr = hd + (size_t)n * heads * Cc + h * Cc;
    const float* av = As + h * Cc;
    const float* bv = Ad + h * Cc;
    float s1 = 0.f, s2 = 0.f;
#pragma unroll 4
    for (int c = 0; c < Cc; ++c) { s1 += hr[c] * av[c]; s2 += dr[c] * bv[c]; }
    asrc[t] = s1;
    adst[t] = s2;
}

__device__ __forceinline__ float edge_logit(const float* asrc, const float* adst,
                                            const float* ce, int s, int d,
                                            float w, int h, int heads)
{
    float a = asrc[s * heads + h] + adst[d * heads + h] + w * ce[h];
    return a > 0.f ? a : 0.2f * a;      // leaky_relu(a, 0.2)
}

// ---------------- pass 1: segment max of logits over dst --------------------
__global__ void edge_pass1(const int* __restrict__ src, const int* __restrict__ dst,
                           const float* __restrict__ ew,
                           const float* __restrict__ asrc, const float* __restrict__ adst,
                           const float* __restrict__ ce, unsigned* __restrict__ m,
                           int n_edges, int heads)
{
    int e = blockIdx.x * blockDim.x + threadIdx.x;
    if (e >= n_edges) return;
    int s = src[e], d = dst[e];
    float w = ew[e];
    for (int h = 0; h < heads; ++h) {
        float a = edge_logit(asrc, adst, ce, s, d, w, h, heads);
        atomicMax(&m[d * heads + h], fp_order(a));
    }
}

// ---------------- pass 2: denom = segment sum of exp(a - m[dst]) ------------
__global__ void edge_pass2(const int* __restrict__ src, const int* __restrict__ dst,
                           const float* __restrict__ ew,
                           const float* __restrict__ asrc, const float* __restrict__ adst,
                           const float* __restrict__ ce, const unsigned* __restrict__ m,
                           float* __restrict__ denom, int n_edges, int heads)
{
    int e = blockIdx.x * blockDim.x + threadIdx.x;
    if (e >= n_edges) return;
    int s = src[e], d = dst[e];
    float w = ew[e];
    for (int h = 0; h < heads; ++h) {
        float a  = edge_logit(asrc, adst, ce, s, d, w, h, heads);
        float mm = fp_unorder(m[d * heads + h]);
        atomicAdd(&denom[d * heads + h], __expf(a - mm));
    }
}

// ---------------- pass 3: out[dst] += hs[src] * alpha  (wave per edge) ------
__global__ __launch_bounds__(256) void edge_pass3(
    const int* __restrict__ src, const int* __restrict__ dst,
    const float* __restrict__ ew,
    const float* __restrict__ asrc, const float* __restrict__ adst,
    const float* __restrict__ ce, const unsigned* __restrict__ m,
    const float* __restrict__ denom, const float* __restrict__ hs,
    float* __restrict__ out, int n_edges, int heads, int F)
{
    int e    = (blockIdx.x * blockDim.x + threadIdx.x) >> 5;   // wave id == edge
    int lane = threadIdx.x & 31;
    if (e >= n_edges) return;
    int s = src[e], d = dst[e];
    float w = ew[e];

    float alpha[Hh];
    for (int h = 0; h < heads; ++h) {
        float a  = edge_logit(asrc, adst, ce, s, d, w, h, heads);
        float mm = fp_unorder(m[d * heads + h]);
        alpha[h] = __expf(a - mm) / (denom[d * heads + h] + 1e-16f);
    }

    const float* hrow = hs + (size_t)s * F;
    float*       orow = out + (size_t)d * F;
    for (int f = lane; f < F; f += 32)
        atomicAdd(&orow[f], hrow[f] * alpha[f >> 6]);   // C == 64
}

// ---------------- epilogue: acc = (acc + b) [relu] --------------------------
__global__ void finalize(float* __restrict__ acc, const float* __restrict__ b,
                         int n_nodes, int F, int do_relu)
{
    int i = blockIdx.x * blockDim.x + threadIdx.x;
    if (i >= n_nodes * F) return;
    float v = acc[i] + b[i % F];
    if (do_relu) v = fmaxf(v, 0.f);
    acc[i] = v;
}

// ---------------------------------------------------------------------------
extern "C" void kernel_launch(void* const* d_in, const int* in_sizes, int n_in,
                              void* d_out, int out_size, void* d_ws, size_t ws_size,
                              hipStream_t stream)
{
    const float* x    = (const float*)d_in[0];
    const int*   ei   = (const int*)d_in[1];
    const int*   srcE = ei;
    const int*   dstE = ei + Ee;
    const float* ew   = (const float*)d_in[2];
    const float* w1s  = (const float*)d_in[3];
    const float* w1d  = (const float*)d_in[4];
    const float* a1s  = (const float*)d_in[5];
    const float* a1d  = (const float*)d_in[6];
    const float* w1e  = (const float*)d_in[7];
    const float* a1e  = (const float*)d_in[8];
    const float* b1   = (const float*)d_in[9];
    const float* w2   = (const float*)d_in[10];
    const float* a2s  = (const float*)d_in[11];
    const float* a2d  = (const float*)d_in[12];
    const float* w2e  = (const float*)d_in[13];
    const float* a2e  = (const float*)d_in[14];
    const float* b2   = (const float*)d_in[15];
    const float* w3   = (const float*)d_in[16];
    const float* a3s  = (const float*)d_in[17];
    const float* a3d  = (const float*)d_in[18];
    const float* w3e  = (const float*)d_in[19];
    const float* a3e  = (const float*)d_in[20];
    const float* b3   = (const float*)d_in[21];
    float* out = (float*)d_out;

    // ---- workspace carve-up (floats) ----
    float*    ws  = (float*)d_ws;
    float*    hsb = ws;                                   // N*192 projections (src)
    float*    hdb = hsb + (size_t)Nn_NODES * FHID;        // N*192 projections (dst) / acc
    float*    ob  = hdb + (size_t)Nn_NODES * FHID;        // N*192 accumulator / hidden
    float*    as_ = ob  + (size_t)Nn_NODES * FHID;        // N*H
    float*    ad_ = as_ + (size_t)Nn_NODES * Hh;          // N*H
    unsigned* mb  = (unsigned*)(ad_ + (size_t)Nn_NODES * Hh); // N*H
    float*    dn  = (float*)(mb + (size_t)Nn_NODES * Hh); // N*H
    float*    ce  = dn + (size_t)Nn_NODES * Hh;           // 8

    const int TB = 256;
    const int edge_blocks  = (Ee + TB - 1) / TB;
    const int wave_blocks  = (Ee * 32) / TB;              // pass3: wave per edge
    const int nodeF_blocks = (Nn_NODES * FHID + TB - 1) / TB;
    const int nodeO_blocks = (Nn_NODES * OUTF + TB - 1) / TB;
    const int ndot3_blocks = (Nn_NODES * Hh + TB - 1) / TB;
    const int ndot1_blocks = (Nn_NODES + TB - 1) / TB;

    // edge-weight attention scalars for all 3 layers
    ce_kernel<<<1, 32, 0, stream>>>(w1e, a1e, w2e, a2e, w3e, a3e, ce);

    // =================== Layer 1 (bipartite linears, H=3) ===================
    {
        dim3 g(Nn_NODES / 16, FHID / 64);
        gemm_f32_wmma<<<g, 32, 0, stream>>>(x, w1s, hsb, Nn_NODES, F_IN, FHID);
        gemm_f32_wmma<<<g, 32, 0, stream>>>(x, w1d, hdb, Nn_NODES, F_IN, FHID);
    }
    node_dots<<<ndot3_blocks, TB, 0, stream>>>(hsb, hdb, a1s, a1d, as_, ad_, Nn_NODES, Hh);
    hipMemsetAsync(mb, 0, (size_t)Nn_NODES * Hh * 4, stream);
    hipMemsetAsync(dn, 0, (size_t)Nn_NODES * Hh * 4, stream);
    hipMemsetAsync(ob, 0, (size_t)Nn_NODES * FHID * 4, stream);
    edge_pass1<<<edge_blocks, TB, 0, stream>>>(srcE, dstE, ew, as_, ad_, ce + 0, mb, Ee, Hh);
    edge_pass2<<<edge_blocks, TB, 0, stream>>>(srcE, dstE, ew, as_, ad_, ce + 0, mb, dn, Ee, Hh);
    edge_pass3<<<wave_blocks, TB, 0, stream>>>(srcE, dstE, ew, as_, ad_, ce + 0, mb, dn,
                                               hsb, ob, Ee, Hh, FHID);
    finalize<<<nodeF_blocks, TB, 0, stream>>>(ob, b1, Nn_NODES, FHID, 1);   // h1 in ob

    // =================== Layer 2 (shared linear, H=3) =======================
    {
        dim3 g(Nn_NODES / 16, FHID / 64);
        gemm_f32_wmma<<<g, 32, 0, stream>>>(ob, w2, hsb, Nn_NODES, FHID, FHID);
    }
    node_dots<<<ndot3_blocks, TB, 0, stream>>>(hsb, hsb, a2s, a2d, as_, ad_, Nn_NODES, Hh);
    hipMemsetAsync(mb, 0, (size_t)Nn_NODES * Hh * 4, stream);
    hipMemsetAsync(dn, 0, (size_t)Nn_NODES * Hh * 4, stream);
    hipMemsetAsync(hdb, 0, (size_t)Nn_NODES * FHID * 4, stream);
    edge_pass1<<<edge_blocks, TB, 0, stream>>>(srcE, dstE, ew, as_, ad_, ce + 3, mb, Ee, Hh);
    edge_pass2<<<edge_blocks, TB, 0, stream>>>(srcE, dstE, ew, as_, ad_, ce + 3, mb, dn, Ee, Hh);
    edge_pass3<<<wave_blocks, TB, 0, stream>>>(srcE, dstE, ew, as_, ad_, ce + 3, mb, dn,
                                               hsb, hdb, Ee, Hh, FHID);
    finalize<<<nodeF_blocks, TB, 0, stream>>>(hdb, b2, Nn_NODES, FHID, 1);  // h2 in hdb

    // =================== Layer 3 (shared linear, H=1, C=64) =================
    {
        dim3 g(Nn_NODES / 16, OUTF / 64);
        gemm_f32_wmma<<<g, 32, 0, stream>>>(hdb, w3, ob, Nn_NODES, FHID, OUTF); // proj N*64 in ob
    }
    node_dots<<<ndot1_blocks, TB, 0, stream>>>(ob, ob, a3s, a3d, as_, ad_, Nn_NODES, 1);
    hipMemsetAsync(mb, 0, (size_t)Nn_NODES * 4, stream);
    hipMemsetAsync(dn, 0, (size_t)Nn_NODES * 4, stream);
    hipMemsetAsync(out, 0, (size_t)Nn_NODES * OUTF * 4, stream);
    edge_pass1<<<edge_blocks, TB, 0, stream>>>(srcE, dstE, ew, as_, ad_, ce + 6, mb, Ee, 1);
    edge_pass2<<<edge_blocks, TB, 0, stream>>>(srcE, dstE, ew, as_, ad_, ce + 6, mb, dn, Ee, 1);
    edge_pass3<<<wave_blocks, TB, 0, stream>>>(srcE, dstE, ew, as_, ad_, ce + 6, mb, dn,
                                               ob, out, Ee, 1, OUTF);
    finalize<<<nodeO_blocks, TB, 0, stream>>>(out, b3, Nn_NODES, OUTF, 0);  // no relu
}